// Recommender_75917841924564
// MI455X (gfx1250) — compile-verified
//
#include <hip/hip_runtime.h>
#include <hip/hip_bf16.h>

typedef float v2f __attribute__((ext_vector_type(2)));
typedef float v8f __attribute__((ext_vector_type(8)));

#define N_ENT   100000
#define DIM     64
#define N_REL   16
#define N_EDGES 2000000
#define N_ITEMS 30000
#define BATCH   4096
#define N_HOPS  3
#define DECAY_F 1e-4f

// ---------------------------------------------------------------------------
// Fused item attention: out_x = pic@W_PF+b, out_y = txt@W_TF+b,
// a=relu(out_x@W_w), b=relu(out_y@W_w), softmax over {a,b}, combine.
// One wave32 per 16-item block, f32 WMMA 16x16x4 on the matrix pipe.
// ---------------------------------------------------------------------------
__launch_bounds__(32)
__global__ void item_attention_kernel(const float* __restrict__ picture,
                                      const float* __restrict__ text,
                                      const float* __restrict__ W_PF,
                                      const float* __restrict__ b_PF,
                                      const float* __restrict__ W_TF,
                                      const float* __restrict__ b_TF,
                                      const float* __restrict__ W_w,
                                      float* __restrict__ item_emb) {
    __shared__ __align__(16) float ldsx[16][DIM];
    __shared__ __align__(16) float ldsy[16][DIM];
    __shared__ float cw[2][16];

    const int lane = threadIdx.x;     // 0..31
    const int m    = lane & 15;       // row within 16-block (A), col n (B/C)
    const int kh   = lane >> 4;       // K-half select
    const int itemBase = blockIdx.x * 16;

    v8f accx[4] = {};   // 16x64 f32 accumulators (4 N-tiles of 16)
    v8f accy[4] = {};

    // ---- picture @ W_PF  (K = 2048) ----
    {
        const float* Arow = picture + (size_t)(itemBase + m) * 2048;
        for (int k0 = 0; k0 < 2048; k0 += 4) {
            const int ka = k0 + 2 * kh;
            v2f a = *(const v2f*)(Arow + ka);           // A[m][ka], A[m][ka+1]
#pragma unroll
            for (int t = 0; t < 4; ++t) {
                v2f b;
                b.x = W_PF[(size_t)ka * 64 + t * 16 + m];
                b.y = W_PF[(size_t)(ka + 1) * 64 + t * 16 + m];
                accx[t] = __builtin_amdgcn_wmma_f32_16x16x4_f32(
                    false, a, false, b, (short)0, accx[t], false, false);
            }
        }
    }
    // ---- text @ W_TF  (K = 768) ----
    {
        const float* Arow = text + (size_t)(itemBase + m) * 768;
        for (int k0 = 0; k0 < 768; k0 += 4) {
            const int ka = k0 + 2 * kh;
            v2f a = *(const v2f*)(Arow + ka);
#pragma unroll
            for (int t = 0; t < 4; ++t) {
                v2f b;
                b.x = W_TF[(size_t)ka * 64 + t * 16 + m];
                b.y = W_TF[(size_t)(ka + 1) * 64 + t * 16 + m];
                accy[t] = __builtin_amdgcn_wmma_f32_16x16x4_f32(
                    false, a, false, b, (short)0, accy[t], false, false);
            }
        }
    }

    // C/D layout: VGPR v -> row (v + 8*kh), col = t*16 + (lane&15)
#pragma unroll
    for (int t = 0; t < 4; ++t) {
        const int col = t * 16 + m;
        const float bx = b_PF[col];
        const float by = b_TF[col];
#pragma unroll
        for (int v = 0; v < 8; ++v) {
            const int row = v + 8 * kh;
            ldsx[row][col] = accx[t][v] + bx;
            ldsy[row][col] = accy[t][v] + by;
        }
    }
    __syncthreads();

    // per-row attention logits: relu(row . W_w)
    if (lane < 16) {
        float s = 0.0f;
        for (int d = 0; d < DIM; ++d) s += ldsx[lane][d] * W_w[d];
        cw[0][lane] = fmaxf(s, 0.0f);
    } else {
        const int r = lane - 16;
        float s = 0.0f;
        for (int d = 0; d < DIM; ++d) s += ldsy[r][d] * W_w[d];
        cw[1][r] = fmaxf(s, 0.0f);
    }
    __syncthreads();

    // combine: lane -> row = lane>>1, half-row of 32 cols
    const int row   = lane >> 1;
    const int cbase = (lane & 1) * 32;
    const float ea  = expf(cw[0][row]);
    const float eb  = expf(cw[1][row]);
    const float inv = 1.0f / (ea + eb);
    const float cx = ea * inv, cy = eb * inv;
    float* out = item_emb + (size_t)(itemBase + row) * DIM + cbase;
#pragma unroll
    for (int c = 0; c < 32; c += 4) {
        float4 x = *(const float4*)&ldsx[row][cbase + c];
        float4 y = *(const float4*)&ldsy[row][cbase + c];
        float4 o;
        o.x = cx * x.x + cy * y.x;
        o.y = cx * x.y + cy * y.y;
        o.z = cx * x.z + cy * y.z;
        o.w = cx * x.w + cy * y.w;
        *(float4*)(out + c) = o;
    }
}

// ---------------------------------------------------------------------------
// Graph conv pieces
// ---------------------------------------------------------------------------
__global__ void degree_kernel(const int* __restrict__ head,
                              float* __restrict__ counts) {
    int e = blockIdx.x * blockDim.x + threadIdx.x;
    if (e < N_EDGES) atomicAdd(&counts[head[e]], 1.0f);
}

__global__ void init_kernel(const float* __restrict__ src,
                            float* __restrict__ e_cur,
                            float* __restrict__ res) {
    int i = blockIdx.x * blockDim.x + threadIdx.x;
    if (i < N_ENT * DIM) {
        float v = src[i];
        e_cur[i] = v;
        res[i] = v;
    }
}

// 16 lanes per edge, float4 per lane: msg = e[tail]*rel; agg[head] += msg
__global__ void scatter_kernel(const float* __restrict__ e_cur,
                               const int* __restrict__ head,
                               const int* __restrict__ tail,
                               const int* __restrict__ etype,
                               const float* __restrict__ weight,
                               float* __restrict__ agg) {
    long long gid = (long long)blockIdx.x * blockDim.x + threadIdx.x;
    if (gid >= (long long)N_EDGES * 16) return;
    const int e = (int)(gid >> 4);
    const int g = ((int)gid & 15) * 4;
    const int h = head[e];
    const int t = tail[e];
    const int r = etype[e] - 1;
    const float4 ev = *(const float4*)(e_cur + (size_t)t * DIM + g);
    const float4 rv = *(const float4*)(weight + (size_t)r * DIM + g);
    float* dst = agg + (size_t)h * DIM + g;
    atomicAdd(dst + 0, ev.x * rv.x);
    atomicAdd(dst + 1, ev.y * rv.y);
    atomicAdd(dst + 2, ev.z * rv.z);
    atomicAdd(dst + 3, ev.w * rv.w);
}

// one wave32 per entity: agg/denom, L2-normalize row, res += e_new
__global__ void normalize_kernel(const float* __restrict__ agg,
                                 const float* __restrict__ counts,
                                 float* __restrict__ e_out,
                                 float* __restrict__ res) {
    const int wid  = (blockIdx.x * blockDim.x + threadIdx.x) >> 5;
    const int lane = threadIdx.x & 31;
    if (wid >= N_ENT) return;
    const float den = fmaxf(counts[wid], 1.0f);
    const size_t base = (size_t)wid * DIM;
    float v0 = agg[base + lane] / den;
    float v1 = agg[base + lane + 32] / den;
    float s = v0 * v0 + v1 * v1;
#pragma unroll
    for (int msk = 16; msk > 0; msk >>= 1) s += __shfl_xor(s, msk, 32);
    const float inv = 1.0f / fmaxf(sqrtf(s), 1e-12f);
    v0 *= inv;
    v1 *= inv;
    e_out[base + lane]      = v0;
    e_out[base + lane + 32] = v1;
    res[base + lane]        += v0;
    res[base + lane + 32]   += v1;
}

// ---------------------------------------------------------------------------
// BPR loss + regularizer
// ---------------------------------------------------------------------------
__global__ void loss_kernel(const float* __restrict__ res,
                            const float* __restrict__ item_emb,
                            const int* __restrict__ users,
                            const int* __restrict__ pos,
                            const int* __restrict__ neg,
                            float* __restrict__ acc) {
    __shared__ float smf[256];
    __shared__ float srg[256];
    const int i = blockIdx.x * blockDim.x + threadIdx.x;
    float mf = 0.0f, rg = 0.0f;
    if (i < BATCH) {
        const float* u = res + (size_t)users[i] * DIM;
        const float* p = item_emb + (size_t)pos[i] * DIM;
        const float* n = item_emb + (size_t)neg[i] * DIM;
        float ps = 0, ns = 0, su = 0, sp = 0, sn = 0;
        for (int d = 0; d < DIM; ++d) {
            const float uv = u[d], pv = p[d], nv = n[d];
            ps += uv * pv;
            ns += uv * nv;
            su += uv * uv;
            sp += pv * pv;
            sn += nv * nv;
        }
        const float z = ns - ps;  // -(pos_s - neg_s)
        // -log_sigmoid(x) = softplus(-x), numerically stable
        mf = (z > 0.0f) ? (z + log1pf(expf(-z))) : log1pf(expf(z));
        rg = su + sp + sn;
    }
    smf[threadIdx.x] = mf;
    srg[threadIdx.x] = rg;
    __syncthreads();
    for (int st = 128; st > 0; st >>= 1) {
        if (threadIdx.x < st) {
            smf[threadIdx.x] += smf[threadIdx.x + st];
            srg[threadIdx.x] += srg[threadIdx.x + st];
        }
        __syncthreads();
    }
    if (threadIdx.x == 0) {
        atomicAdd(&acc[0], smf[0]);
        atomicAdd(&acc[1], srg[0]);
    }
}

__global__ void finalize_kernel(const float* __restrict__ acc,
                                float* __restrict__ out) {
    if (threadIdx.x == 0 && blockIdx.x == 0) {
        const float mf  = acc[0] / (float)BATCH;
        const float emb = DECAY_F * (acc[1] * 0.5f) / (float)BATCH;
        out[0] = mf + emb;
        out[1] = mf;
        out[2] = emb;
    }
}

// ---------------------------------------------------------------------------
extern "C" void kernel_launch(void* const* d_in, const int* in_sizes, int n_in,
                              void* d_out, int out_size, void* d_ws, size_t ws_size,
                              hipStream_t stream) {
    const float* entity_emb = (const float*)d_in[0];
    const float* weight     = (const float*)d_in[1];
    const float* picture    = (const float*)d_in[2];
    const float* text       = (const float*)d_in[3];
    const float* W_PF       = (const float*)d_in[4];
    const float* b_PF       = (const float*)d_in[5];
    const float* W_TF       = (const float*)d_in[6];
    const float* b_TF       = (const float*)d_in[7];
    const float* W_w        = (const float*)d_in[8];
    const int*   edge_index = (const int*)d_in[9];
    const int*   head       = edge_index;
    const int*   tail       = edge_index + N_EDGES;
    const int*   etype      = (const int*)d_in[10];
    const int*   users      = (const int*)d_in[11];
    const int*   pos        = (const int*)d_in[12];
    const int*   neg        = (const int*)d_in[13];
    float* out = (float*)d_out;

    // workspace carve-up
    char* ws = (char*)d_ws;
    float* e_cur    = (float*)ws; ws += sizeof(float) * (size_t)N_ENT * DIM;
    float* agg      = (float*)ws; ws += sizeof(float) * (size_t)N_ENT * DIM;
    float* res      = (float*)ws; ws += sizeof(float) * (size_t)N_ENT * DIM;
    float* item_emb = (float*)ws; ws += sizeof(float) * (size_t)N_ITEMS * DIM;
    float* counts   = (float*)ws; ws += sizeof(float) * (size_t)N_ENT;
    float* acc      = (float*)ws; ws += sizeof(float) * 2;

    // 1) fused multimodal item embedding (WMMA f32)
    item_attention_kernel<<<N_ITEMS / 16, 32, 0, stream>>>(
        picture, text, W_PF, b_PF, W_TF, b_TF, W_w, item_emb);

    // 2) degrees (denominator) once
    hipMemsetAsync(counts, 0, sizeof(float) * (size_t)N_ENT, stream);
    degree_kernel<<<(N_EDGES + 255) / 256, 256, 0, stream>>>(head, counts);

    // 3) init e/res
    init_kernel<<<(N_ENT * DIM + 255) / 256, 256, 0, stream>>>(entity_emb, e_cur, res);

    // 4) 3-hop relational conv (gather * rel -> scatter-mean -> normalize -> accumulate)
    for (int hop = 0; hop < N_HOPS; ++hop) {
        hipMemsetAsync(agg, 0, sizeof(float) * (size_t)N_ENT * DIM, stream);
        scatter_kernel<<<(int)(((long long)N_EDGES * 16 + 255) / 256), 256, 0, stream>>>(
            e_cur, head, tail, etype, weight, agg);
        normalize_kernel<<<(N_ENT * 32 + 255) / 256, 256, 0, stream>>>(
            agg, counts, e_cur, res);
    }

    // 5) loss
    hipMemsetAsync(acc, 0, sizeof(float) * 2, stream);
    loss_kernel<<<(BATCH + 255) / 256, 256, 0, stream>>>(res, item_emb, users, pos, neg, acc);
    finalize_kernel<<<1, 1, 0, stream>>>(acc, out);
}